// MultiHeadSelfAttention_15178414424602
// MI455X (gfx1250) — compile-verified
//
#include <hip/hip_runtime.h>
#include <math.h>

typedef float v2f __attribute__((ext_vector_type(2)));
typedef float v8f __attribute__((ext_vector_type(8)));

// D = A(16x4) * B(4x16) + C, full f32 on the matrix unit.
// 8-arg form: (neg_a, A, neg_b, B, c_mod, C, reuse_a, reuse_b)
#define WMMA_F32(a, b, c) \
  __builtin_amdgcn_wmma_f32_16x16x4_f32(false, (a), false, (b), (short)0, (c), false, false)

// ---------------------------------------------------------------------------
// Generic f32 GEMM with bias:  C[M,N] = A[M,K] * B[K,N] + bias[N]
// Block tile 128x64, BK=16, 256 threads = 8 waves (4x2), 2x2 WMMA tiles/wave.
// ---------------------------------------------------------------------------
__global__ __launch_bounds__(256)
void gemm_bias_f32_wmma(const float* __restrict__ A, const float* __restrict__ Bm,
                        const float* __restrict__ bias, float* __restrict__ C,
                        int M, int N, int K)
{
    constexpr int BM = 128, BN = 64, BK = 16;
    constexpr int ALD = BK + 2;               // pad to spread LDS banks
    __shared__ float As[BM * ALD];            // 9.2 KB
    __shared__ float Bs[BK * BN];             // 4.0 KB

    const int tid  = threadIdx.x;
    const int lane = tid & 31;
    const int wave = tid >> 5;
    const int wm   = wave & 3;                // 0..3 -> M offset
    const int wn   = wave >> 2;               // 0..1 -> N offset
    const int m0   = blockIdx.y * BM;
    const int n0   = blockIdx.x * BN;
    const int fm   = lane & 15;               // fragment row/col within tile
    const int kq   = lane >> 4;               // K-pair selector (0/1)

    v8f acc[2][2] = {};

    for (int kc = 0; kc < K; kc += BK) {
        // cooperative load of A tile (2048 floats)
        #pragma unroll
        for (int i = 0; i < 8; ++i) {
            int e = tid + i * 256;
            int m = e >> 4, k = e & 15;
            As[m * ALD + k] = A[(size_t)(m0 + m) * K + kc + k];
        }
        // cooperative load of B tile (1024 floats)
        #pragma unroll
        for (int i = 0; i < 4; ++i) {
            int e = tid + i * 256;
            int k = e >> 6, n = e & 63;
            Bs[k * BN + n] = Bm[(size_t)(kc + k) * N + n0 + n];
        }
        __syncthreads();

        #pragma unroll
        for (int kb = 0; kb < BK; kb += 4) {
            v2f af[2], bf[2];
            #pragma unroll
            for (int mi = 0; mi < 2; ++mi) {
                int mrow = wm * 32 + mi * 16 + fm;
                af[mi][0] = As[mrow * ALD + kb + 2 * kq + 0];
                af[mi][1] = As[mrow * ALD + kb + 2 * kq + 1];
            }
            #pragma unroll
            for (int ni = 0; ni < 2; ++ni) {
                int ncol = wn * 32 + ni * 16 + fm;
                bf[ni][0] = Bs[(kb + 2 * kq + 0) * BN + ncol];
                bf[ni][1] = Bs[(kb + 2 * kq + 1) * BN + ncol];
            }
            #pragma unroll
            for (int mi = 0; mi < 2; ++mi)
                #pragma unroll
                for (int ni = 0; ni < 2; ++ni)
                    acc[mi][ni] = WMMA_F32(af[mi], bf[ni], acc[mi][ni]);
        }
        __syncthreads();
    }

    // epilogue: bias add + store. C/D layout: M = kq*8 + v, N = fm.
    #pragma unroll
    for (int mi = 0; mi < 2; ++mi)
        #pragma unroll
        for (int ni = 0; ni < 2; ++ni) {
            int row = m0 + wm * 32 + mi * 16 + kq * 8;
            int col = n0 + wn * 32 + ni * 16 + fm;
            float bv = bias[col];
            #pragma unroll
            for (int v = 0; v < 8; ++v)
                C[(size_t)(row + v) * N + col] = acc[mi][ni][v] + bv;
        }
}

// ---------------------------------------------------------------------------
// Fused attention: one block per (b, h, 16-row q tile).
//   S = (Q K^T)/8 + rel_bias, window mask |i-j|<=128 -> -inf
//   P = softmax(S) -> written to attn output (full 512 cols; masked cols = 0)
//   ctx = P V     -> written to ws (as [B,T,D] with head-major columns)
// ---------------------------------------------------------------------------
__global__ __launch_bounds__(256)
void attn_fused(const float* __restrict__ qkv, const float* __restrict__ rel,
                float* __restrict__ attn_out, float* __restrict__ ctx)
{
    constexpr int Tt = 512, Dm = 1024, Hh = 16, DKk = 64;
    __shared__ float Qs[16 * 66];     // Q tile, padded (4.2 KB)
    __shared__ float Ss[16 * 512];    // scores / probs (32 KB)

    const int bid = blockIdx.x;
    const int qt  = bid & 31;
    const int h   = (bid >> 5) & 15;
    const int b   = bid >> 9;
    const int q0  = qt * 16;

    const int tid  = threadIdx.x;
    const int lane = tid & 31;
    const int wave = tid >> 5;
    const int fm   = lane & 15;
    const int kq   = lane >> 4;

    const size_t rs = 3 * Dm;                 // qkv row stride
    const float* qbase = qkv + (size_t)b * Tt * rs + h * DKk;
    const float* kbase = qbase + Dm;
    const float* vbase = qbase + 2 * Dm;

    // stage Q tile, pre-scaled by 1/sqrt(DK)
    for (int i = tid; i < 16 * 64; i += 256) {
        int r = i >> 6, d = i & 63;
        Qs[r * 66 + d] = qbase[(size_t)(q0 + r) * rs + d] * 0.125f;
    }
    __syncthreads();

    // ---- phase 1: scores via WMMA (M=16, N=512 in 32 j-tiles, K=64) ----
    for (int jt = wave; jt < 32; jt += 8) {
        int j0 = jt * 16;
        v8f acc = {};
        #pragma unroll
        for (int kb = 0; kb < 64; kb += 4) {
            int kd = kb + 2 * kq;
            v2f af, bf;
            af[0] = Qs[fm * 66 + kd + 0];
            af[1] = Qs[fm * 66 + kd + 1];
            bf[0] = kbase[(size_t)(j0 + fm) * rs + kd + 0];
            bf[1] = kbase[(size_t)(j0 + fm) * rs + kd + 1];
            acc = WMMA_F32(af, bf, acc);
        }
        int j = j0 + fm;
        #pragma unroll
        for (int v = 0; v < 8; ++v) {
            int i  = kq * 8 + v;              // local row
            int d  = j - (q0 + i);            // relative position
            float s;
            if (d > 128 || d < -128) s = -INFINITY;
            else                     s = acc[v] + rel[(size_t)(d + 511) * Hh + h];
            Ss[i * 512 + j] = s;
        }
    }
    __syncthreads();

    // ---- phase 2: softmax per row (2 rows/wave), write attn ----
    {
        float* arow = attn_out + (((size_t)b * Hh + h) * Tt + q0) * Tt;
        for (int r = wave * 2; r < wave * 2 + 2; ++r) {
            float m = -INFINITY;
            for (int c = lane; c < 512; c += 32) m = fmaxf(m, Ss[r * 512 + c]);
            #pragma unroll
            for (int off = 16; off > 0; off >>= 1) m = fmaxf(m, __shfl_xor(m, off, 32));
            float sum = 0.f;
            for (int c = lane; c < 512; c += 32) {
                float e = __expf(Ss[r * 512 + c] - m);
                Ss[r * 512 + c] = e;
                sum += e;
            }
            #pragma unroll
            for (int off = 16; off > 0; off >>= 1) sum += __shfl_xor(sum, off, 32);
            float inv = 1.0f / sum;
            for (int c = lane; c < 512; c += 32) {
                float p = Ss[r * 512 + c] * inv;
                Ss[r * 512 + c] = p;
                arow[(size_t)r * Tt + c] = p;   // masked cols are exactly 0
            }
        }
    }
    __syncthreads();

    // ---- phase 3: ctx = P @ V (M=16, N=64, K=512); wave-uniform branch ----
    if (wave < 4) {
        int n0 = wave * 16;
        v8f acc = {};
        for (int kb = 0; kb < 512; kb += 4) {
            int kk = kb + 2 * kq;
            v2f af, bf;
            af[0] = Ss[fm * 512 + kk + 0];
            af[1] = Ss[fm * 512 + kk + 1];
            bf[0] = vbase[(size_t)(kk + 0) * rs + n0 + fm];
            bf[1] = vbase[(size_t)(kk + 1) * rs + n0 + fm];
            acc = WMMA_F32(af, bf, acc);
        }
        #pragma unroll
        for (int v = 0; v < 8; ++v) {
            int i = kq * 8 + v;
            ctx[((size_t)(b * Tt) + q0 + i) * Dm + h * DKk + n0 + fm] = acc[v];
        }
    }
}

// ---------------------------------------------------------------------------
extern "C" void kernel_launch(void* const* d_in, const int* in_sizes, int n_in,
                              void* d_out, int out_size, void* d_ws, size_t ws_size,
                              hipStream_t stream)
{
    const float* x    = (const float*)d_in[0];   // [16,512,1024]
    const float* Wqkv = (const float*)d_in[1];   // [1024,3072]
    const float* bqkv = (const float*)d_in[2];   // [3072]
    const float* Wo   = (const float*)d_in[3];   // [1024,1024]
    const float* bo   = (const float*)d_in[4];   // [1024]
    const float* rel  = (const float*)d_in[5];   // [1023,16]

    float* out  = (float*)d_out;                         // [16,512,1024]
    float* attn = out + (size_t)16 * 512 * 1024;         // [16,16,512,512]

    float* qkv = (float*)d_ws;                           // [16,512,3072]  100.7 MB
    float* ctx = qkv + (size_t)16 * 512 * 3072;          // [16,512,1024]   33.6 MB

    // 1) qkv = x @ Wqkv + bqkv   (M=8192, N=3072, K=1024)
    gemm_bias_f32_wmma<<<dim3(3072 / 64, 8192 / 128), 256, 0, stream>>>(
        x, Wqkv, bqkv, qkv, 8192, 3072, 1024);

    // 2) fused scores + softmax + attn-write + ctx   (16*16*32 = 8192 blocks)
    attn_fused<<<dim3(16 * 16 * 32), 256, 0, stream>>>(qkv, rel, attn, ctx);

    // 3) out = ctx @ Wo + bo     (M=8192, N=1024, K=1024)
    gemm_bias_f32_wmma<<<dim3(1024 / 64, 8192 / 128), 256, 0, stream>>>(
        ctx, Wo, bo, out, 8192, 1024, 1024);
}